// GATPrimitive_AttentionScore_41807211659467
// MI455X (gfx1250) — compile-verified
//
#include <hip/hip_runtime.h>

// GAT attention-score primitive for MI455X (gfx1250, wave32).
// logits[e] = dot(Wx_i[e,:], a[0:128]) + dot(Wx_j[e,:], a[128:256]); leaky_relu(0.2)
//
// Memory-bound: 655 MB streamed once -> ~28us floor at 23.3 TB/s. We use
// non-temporal loads (stream > 192MB L2) and V_WMMA_F32_16X16X4_F32 to do the
// 16-row GEMV tiles (B = attention vector broadcast along N, so column 0 of D
// holds the 16 dot products).

typedef float v2f __attribute__((ext_vector_type(2)));
typedef float v8f __attribute__((ext_vector_type(8)));

#define E_ROWS   640000
#define DIM      128
#define TILES    (E_ROWS / 16)        // 40000
#define WAVES_PB 8                    // 256 threads / wave32
#define NEG_SLOPE 0.2f

__global__ __launch_bounds__(256) void gat_score_kernel(
    const float* __restrict__ Wx_i,
    const float* __restrict__ Wx_j,
    const float* __restrict__ a_vec,
    float* __restrict__ out)
{
    // Attention vector (256 floats = 1KB) staged in LDS; every WMMA step
    // reads its B fragment from here with a ds_load_b64.
    __shared__ float a_lds[2 * DIM];
    a_lds[threadIdx.x] = a_vec[threadIdx.x];
    __syncthreads();

    const int lane = threadIdx.x & 31;
    const int wave = threadIdx.x >> 5;
    const int tile = blockIdx.x * WAVES_PB + wave;   // one 16-row tile per wave
    if (tile >= TILES) return;                        // grid is exact; never taken

    const int m = lane & 15;   // row within tile (A-matrix M)
    const int h = lane >> 4;   // K-half select: lanes 0-15 hold K={0,1}, 16-31 hold K={2,3}

    // A-fragment layout for V_WMMA_F32_16X16X4_F32 (ISA 7.12.2):
    //   v0 = W[m][kb + 2h], v1 = W[m][kb + 2h + 1]  -> contiguous b64 per lane.
    const float* __restrict__ pi = Wx_i + (size_t)(tile * 16 + m) * DIM + 2 * h;
    const float* __restrict__ pj = Wx_j + (size_t)(tile * 16 + m) * DIM + 2 * h;
    // B-fragment (4x16, broadcast along N): lane needs (a[kb+2h], a[kb+2h+1]).
    const float* bs = &a_lds[2 * h];          // a_src half
    const float* bd = &a_lds[DIM + 2 * h];    // a_dst half

    v8f ci = {};   // accumulator chain for Wx_i @ a_src
    v8f cj = {};   // accumulator chain for Wx_j @ a_dst  (independent WMMA chain)

#pragma unroll
    for (int k = 0; k < DIM; k += 4) {
        // Streamed weights: non-temporal (NT) so 655MB doesn't thrash L2.
        v2f ai = __builtin_nontemporal_load((const v2f*)(pi + k));
        v2f aj = __builtin_nontemporal_load((const v2f*)(pj + k));
        v2f bi = *(const v2f*)(bs + k);
        v2f bj = *(const v2f*)(bd + k);
        ci = __builtin_amdgcn_wmma_f32_16x16x4_f32(
                 false, ai, false, bi, (short)0, ci, false, false);
        cj = __builtin_amdgcn_wmma_f32_16x16x4_f32(
                 false, aj, false, bj, (short)0, cj, false, false);
    }

    v8f c = ci + cj;

    // C/D 16x16 f32 layout: lane with N==0 holds M = r (lane 0) / M = 8+r (lane 16)
    // in VGPR r. Those two lanes write the 16 logits (leaky-ReLU applied).
    if (m == 0) {
        float* o = out + (size_t)tile * 16 + h * 8;
#pragma unroll
        for (int r = 0; r < 8; ++r) {
            float x = c[r];
            o[r] = x > 0.0f ? x : NEG_SLOPE * x;
        }
    }
}

extern "C" void kernel_launch(void* const* d_in, const int* in_sizes, int n_in,
                              void* d_out, int out_size, void* d_ws, size_t ws_size,
                              hipStream_t stream) {
    const float* Wx_i = (const float*)d_in[0];
    const float* Wx_j = (const float*)d_in[1];
    const float* a    = (const float*)d_in[2];
    float* out        = (float*)d_out;

    dim3 block(256);                       // 8 wave32 waves
    dim3 grid(TILES / WAVES_PB);           // 5000 blocks -> 40000 waves, 1 tile each
    hipLaunchKernelGGL(gat_score_kernel, grid, block, 0, stream,
                       Wx_i, Wx_j, a, out);
}